// RecursiveNonlinearEquationSolver_55774445306377
// MI455X (gfx1250) — compile-verified
//
#include <hip/hip_runtime.h>
#include <hip/hip_bf16.h>

// CDNA5 / gfx1250: wave32, WMMA f32 16x16x4.
typedef float v2f __attribute__((ext_vector_type(2)));
typedef float v8f __attribute__((ext_vector_type(8)));

namespace {
constexpr float kDT = 0.01f;
constexpr int kS = 16;       // nstate
constexpr int kB = 256;      // batch
constexpr int kN = 512;      // time steps
constexpr int kChunk = 64;   // BLOCK
constexpr int kNewton = 4;   // NEWTON_ITERS
constexpr int kNeumann = 3;  // Neumann terms for Jd^{-1} (||M||~1e-2 -> ~1e-8 error)
}

// Full 16x16x16 f32 matmul as 4 chained K=4 WMMAs. D = A(16x16) * B(16x16) + 0.
__device__ __forceinline__ v8f wmma_16x16x16(const v2f A[4], const v2f B[4]) {
  v8f c = {};
#pragma unroll
  for (int q = 0; q < 4; ++q)
    c = __builtin_amdgcn_wmma_f32_16x16x4_f32(false, A[q], false, B[q],
                                              (short)0, c, false, false);
  return c;
}

// Write a C/D-layout register tile (lane: N=lo, M = half*8 + i) into a 16x16
// state-major LDS tile lds[s*16 + n].
__device__ __forceinline__ void scatter_C(float* lds, int half, int lo, v8f v) {
#pragma unroll
  for (int i = 0; i < 8; ++i) lds[(half * 8 + i) * kS + lo] = v[i];
}

// Read B fragments (4x16 per K-chunk) from lds[s*16+n].
// Lanes 0-15: regs hold K = 4q, 4q+1; lanes 16-31: K = 4q+2, 4q+3 (ISA 7.12.2).
__device__ __forceinline__ void gather_B(const float* lds, int half, int lo,
                                         v2f bf[4]) {
#pragma unroll
  for (int q = 0; q < 4; ++q) {
    int k = 4 * q + 2 * half;
    bf[q].x = lds[k * kS + lo];
    bf[q].y = lds[(k + 1) * kS + lo];
  }
}

__global__ __launch_bounds__(32) void rnes_solver_kernel(
    const float* __restrict__ y0, const float* __restrict__ W,
    const float* __restrict__ F, float* __restrict__ OUT) {
  __shared__ float Wl[kS * kS];
  __shared__ float Tl[kS * kS];

  const int lane = threadIdx.x;       // 0..31
  const int half = lane >> 4;         // lane-half selects M/K sub-rows
  const int lo   = lane & 15;         // N (batch col) / M (A row)
  const int b    = blockIdx.x * 16 + lo;          // this wave owns 16 batches
  const size_t rowStride = (size_t)kB * kS;       // floats per time-step row
  const size_t laneOff   = (size_t)b * kS + half * 8;  // 32B-aligned

  // Stage W in LDS, then build constant A fragments (A[m][k] = W[m][k]).
#pragma unroll
  for (int i = 0; i < 8; ++i) Wl[lane * 8 + i] = W[lane * 8 + i];
  __syncthreads();
  v2f A[4];
#pragma unroll
  for (int q = 0; q < 4; ++q) {
    int k = 4 * q + 2 * half;
    A[q].x = Wl[lo * kS + k];
    A[q].y = Wl[lo * kS + k + 1];
  }

  // OUT[0] = y0 (reference prepends y0 to the trajectory).
  v8f yprev = *(const v8f*)(y0 + laneOff);
  *(v8f*)(OUT + laneOff) = yprev;

  const v8f zero = {};
  int k1 = 1;
  while (k1 < kN) {
    const int k2 = (k1 + kChunk < kN) ? (k1 + kChunk) : kN;

    // ZeroPredictor: chunk guess = 0 (OUT rows double as the working buffer).
    for (int row = k1; row < k2; ++row)
      *(v8f*)(OUT + (size_t)row * rowStride + laneOff) = zero;

    const v8f prev = yprev;  // previous chunk's final state, fixed over Newton
    for (int it = 0; it < kNewton; ++it) {
      v8f ykm   = prev;  // y_{k-1} (pre-update value within this Newton iter)
      v8f xprev = zero;  // Thomas carry (L = -I  =>  rhs_k = R_k + x_{k-1})
      for (int row = k1; row < k2; ++row) {
        float* yrow = OUT + (size_t)row * rowStride + laneOff;
        const float* frow = F + (size_t)row * rowStride + laneOff;
        if (row + 1 < k2)
          __builtin_prefetch(
              (const void*)(F + (size_t)(row + 1) * rowStride + laneOff), 0, 1);

        v8f y = *(const v8f*)yrow;
        v8f f = *(const v8f*)frow;

        v8f t;
#pragma unroll
        for (int i = 0; i < 8; ++i) t[i] = tanhf(y[i]);

        // W @ tanh(y) for this 16-batch tile via WMMA.
        __syncthreads();
        scatter_C(Tl, half, lo, t);
        __syncthreads();
        v2f Bf[4];
        gather_B(Tl, half, lo, Bf);
        v8f wv = wmma_16x16x16(A, Bf);

        // r = R + x_{k-1},  R = y - y_{k-1} - DT*(W t + f)
        v8f r  = y - ykm - kDT * (wv + f) + xprev;
        v8f om = 1.0f - t * t;  // sech^2 = 1 - tanh^2

        // Neumann solve of (I - DT*W*diag(om)) x = r :  x <- r + M x
        v8f x = r;
#pragma unroll
        for (int ni = 0; ni < kNeumann; ++ni) {
          v8f v = om * x;
          __syncthreads();
          scatter_C(Tl, half, lo, v);
          __syncthreads();
          gather_B(Tl, half, lo, Bf);
          v8f av = wmma_16x16x16(A, Bf);
          x = r + kDT * av;
        }

        v8f ynew = y - x;       // Newton update
        *(v8f*)yrow = ynew;
        ykm = y;                // old y_k feeds row k+1's residual
        xprev = x;              // Thomas carry
      }
    }
    // Reload final chunk state as next chunk's prev.
    yprev = *(const v8f*)(OUT + (size_t)(k2 - 1) * rowStride + laneOff);
    k1 = k2;
  }
}

extern "C" void kernel_launch(void* const* d_in, const int* in_sizes, int n_in,
                              void* d_out, int out_size, void* d_ws,
                              size_t ws_size, hipStream_t stream) {
  (void)in_sizes; (void)n_in; (void)d_ws; (void)ws_size; (void)out_size;
  const float* y0 = (const float*)d_in[0];  // (256,16) f32
  const float* W  = (const float*)d_in[1];  // (16,16)  f32
  const float* F  = (const float*)d_in[2];  // (512,256,16) f32
  float* OUT = (float*)d_out;               // (512,256,16) f32
  // One wave (32 threads) per 16-batch tile; 16 tiles cover batch=256.
  rnes_solver_kernel<<<kB / 16, 32, 0, stream>>>(y0, W, F, OUT);
}